// SGNREC_51917564674254
// MI455X (gfx1250) — compile-verified
//
#include <hip/hip_runtime.h>
#include <math.h>
#include <stdint.h>

// ---------------------------------------------------------------------------
// SR-GNN style session-rec forward on MI455X (gfx1250, wave32).
// Matrix math: V_WMMA_F32_16X16X4_F32 (fp32-exact, 16x16 C tiles, K=4/instr).
// Data movement: Tensor Data Mover (tensor_load_to_lds, 6-arg clang-23 form)
// double-buffered LDS panel staging, synchronized with TENSORcnt; falls back
// to manual float4 copies if the TDM builtin is unavailable.
// ---------------------------------------------------------------------------

typedef __attribute__((ext_vector_type(2))) float v2f;
typedef __attribute__((ext_vector_type(8))) float v8f;
typedef __attribute__((ext_vector_type(4))) unsigned int u32x4;
typedef __attribute__((ext_vector_type(8))) int i32x8;
typedef __attribute__((ext_vector_type(4))) int i32x4;

#define BB   512
#define SS   64
#define DD   256
#define TWOD 512
#define NN   100000
#define NM1  99999

#if defined(__has_builtin)
#if __has_builtin(__builtin_amdgcn_tensor_load_to_lds) && \
    __has_builtin(__builtin_amdgcn_s_wait_tensorcnt)
#define USE_TDM 1
#endif
#endif

static __device__ __forceinline__ v8f wmma4(v2f a, v2f b, v8f c) {
  // D = A(16x4,f32) * B(4x16,f32) + C(16x16,f32)
  return __builtin_amdgcn_wmma_f32_16x16x4_f32(
      /*neg_a=*/false, a, /*neg_b=*/false, b,
      /*c_mod=*/(short)0, c, /*reuse_a=*/false, /*reuse_b=*/false);
}

#ifdef USE_TDM
// ---------------------------------------------------------------------------
// TDM 2D tile load: rows x cols of f32 from a row-major tensor into LDS.
// Descriptor per CDNA5 ISA ch.8: group0 = {count=1 | lds_addr | global_addr |
// type=2}; group1 = {data_size=4B, tensor_dim0/1, tile_dim0/1,
// tensor_dim0_stride}. Groups 2/3 (and the extra group of the 6-arg builtin)
// are zero/disabled for a 2D tensor. Dynamic LDS is the only LDS in these
// kernels, so its block offset base is 0.
// Rows of the tile beyond tensor_h are zero-filled by TDM OOB semantics.
// ---------------------------------------------------------------------------
static __device__ __forceinline__ void tdm_load_2d(
    unsigned lds_off_bytes, const void* gaddr,
    unsigned tile_w, unsigned tile_h,
    unsigned long long row_stride_elts,
    unsigned tensor_w, unsigned tensor_h) {
  unsigned long long ga = (unsigned long long)(uintptr_t)gaddr;
  u32x4 g0;
  g0[0] = 1u;                                        // count=1, user mode
  g0[1] = lds_off_bytes;                             // lds_addr (bytes)
  g0[2] = (unsigned)(ga & 0xFFFFFFFFu);              // global_addr[31:0]
  g0[3] = (unsigned)((ga >> 32) & 0x1FFFFFFu) |      // global_addr[56:32]
          (2u << 30);                                // type = 2 ("image")
  i32x8 g1;
  g1[0] = (int)(2u << 16);                           // data_size=2 -> 4 bytes
  g1[1] = (int)((tensor_w & 0xFFFFu) << 16);         // tensor_dim0[15:0]
  g1[2] = (int)(((tensor_w >> 16) & 0xFFFFu) |       // tensor_dim0[31:16]
                ((tensor_h & 0xFFFFu) << 16));       // tensor_dim1[15:0]
  g1[3] = (int)(((tensor_h >> 16) & 0xFFFFu) |       // tensor_dim1[31:16]
                ((tile_w & 0xFFFFu) << 16));         // tile_dim0
  g1[4] = (int)(tile_h & 0xFFFFu);                   // tile_dim1 (tile_dim2=0)
  g1[5] = (int)(unsigned)(row_stride_elts & 0xFFFFFFFFull);   // stride[31:0]
  g1[6] = (int)(unsigned)((row_stride_elts >> 32) & 0xFFFFull); // stride[47:32]
  g1[7] = 0;
  i32x4 z4 = {0, 0, 0, 0};
  i32x8 z8 = {0, 0, 0, 0, 0, 0, 0, 0};
  __builtin_amdgcn_tensor_load_to_lds(g0, g1, z4, z4, z8, 0);
}
#endif

// ---------------------------------------------------------------------------
// Kernel 1: per-batch  fin = emb[item[b]];  av = [adj_in@fin , adj_out@fin]
// adj ** LAYERS with LAYERS==1 is elementwise identity -> use adj directly.
// Block = batch b, 256 threads (8 waves). LDS: fin [64][256] f32 = 64KB.
// (Row gather is index-dependent -> stays on per-lane float4 loads.)
// ---------------------------------------------------------------------------
__global__ void __launch_bounds__(256) k_adj_av(
    const float* __restrict__ emb, const float* __restrict__ adj_in,
    const float* __restrict__ adj_out, const int* __restrict__ item,
    float* __restrict__ av) {
  extern __shared__ float fin[];  // [64][256]
  const int b   = blockIdx.x;
  const int tid = threadIdx.x;

  for (int it = 0; it < 16; ++it) {
    int i  = tid + it * 256;
    int s  = i >> 6;
    int c4 = i & 63;
    long long row = item[b * SS + s];
    *(float4*)&fin[s * DD + c4 * 4] =
        *(const float4*)&emb[row * DD + c4 * 4];
  }
  __syncthreads();

  const int w = tid >> 5, l = tid & 31, half = l >> 4, lm = l & 15;
  const float* adjb_in  = adj_in  + (long long)b * SS * SS;
  const float* adjb_out = adj_out + (long long)b * SS * SS;
  float* dst = av + (long long)b * SS * TWOD;

  for (int j = 0; j < 16; ++j) {
    int t  = w * 16 + j;       // 0..127
    int mt = t >> 5;           // 0..3
    int nt = t & 31;           // 0..31 (0..15 -> fs_in, 16..31 -> fs_out)
    const float* adj = (nt >= 16) ? adjb_out : adjb_in;
    int d0 = (nt & 15) * 16;

    v8f acc = {};
    for (int kk = 0; kk < SS; kk += 4) {
      v2f a = *(const v2f*)&adj[(mt * 16 + lm) * SS + kk + half * 2];
      v2f bf;
      bf.x = fin[(kk + half * 2) * DD + d0 + lm];
      bf.y = fin[(kk + half * 2 + 1) * DD + d0 + lm];
      acc = wmma4(a, bf, acc);
    }
    int col = ((nt >= 16) ? DD : 0) + d0 + lm;
    for (int i = 0; i < 8; ++i)
      dst[(mt * 16 + i + 8 * half) * TWOD + col] = acc[i];
  }
}

// ---------------------------------------------------------------------------
// Kernel 2/3: f32 WMMA GEMM  C[M,N] = (relu?)(A[M,K] @ B[K,N])
// Block computes 64x64 C tile; 8 waves x 2 tiles (sharing the A fragment).
// TDM path: ping-pong 64x64 A/B panels (4 x 16KB LDS), wave 0 issues
// tensor_load_to_lds for stage i+1 while all waves run WMMA on stage i;
// s_wait_tensorcnt 2 keeps exactly the just-issued pair in flight.
// Requires M%64==0, N%64==0, K%64==0 (true for all uses here).
// ---------------------------------------------------------------------------
__global__ void __launch_bounds__(256) k_gemm_f32(
    const float* __restrict__ A, const float* __restrict__ Bm,
    float* __restrict__ C, int M, int N, int K, int relu) {
  extern __shared__ float lds[];
  const int m0 = blockIdx.y * 64, n0 = blockIdx.x * 64;
  const int tid = threadIdx.x;
  const int w = tid >> 5, l = tid & 31, half = l >> 4, lm = l & 15;
  const int t0   = w * 2;
  const int trow = t0 >> 2;          // shared M-tile row for both tiles
  const int tc0  = t0 & 3, tc1 = (t0 + 1) & 3;

  v8f acc0 = {}, acc1 = {};

#ifdef USE_TDM
  const int stages = K >> 6;
  if (w == 0) {  // prologue: stage 0 -> buffer 0
    tdm_load_2d(0u,     &A[(long long)m0 * K],  64, 64, (unsigned)K, 64, 64);
    tdm_load_2d(16384u, &Bm[(long long)0 + n0], 64, 64, (unsigned)N, 64, 64);
  }
  for (int i = 0; i < stages; ++i) {
    if (w == 0) {
      if (i + 1 < stages) {
        const long long kb = (long long)(i + 1) * 64;
        const unsigned boff = (unsigned)((i + 1) & 1) * 32768u;
        tdm_load_2d(boff, &A[(long long)m0 * K + kb],
                    64, 64, (unsigned)K, 64, 64);
        tdm_load_2d(boff + 16384u, &Bm[kb * N + n0],
                    64, 64, (unsigned)N, 64, 64);
        __builtin_amdgcn_s_wait_tensorcnt(2);  // stage i landed
      } else {
        __builtin_amdgcn_s_wait_tensorcnt(0);
      }
    }
    __syncthreads();  // stage i visible to all waves
    const float* Ap = lds + (i & 1) * 8192;
    const float* Bp = Ap + 4096;
    for (int kk = 0; kk < 64; kk += 4) {
      v2f a = *(const v2f*)&Ap[(trow * 16 + lm) * 64 + kk + half * 2];
      int kr0 = (kk + half * 2) * 64, kr1 = kr0 + 64;
      v2f b0, b1;
      b0.x = Bp[kr0 + tc0 * 16 + lm];
      b0.y = Bp[kr1 + tc0 * 16 + lm];
      b1.x = Bp[kr0 + tc1 * 16 + lm];
      b1.y = Bp[kr1 + tc1 * 16 + lm];
      acc0 = wmma4(a, b0, acc0);
      acc1 = wmma4(a, b1, acc1);
    }
    __syncthreads();  // done reading buffer (i&1) before it is re-filled
  }
#else
  float* Ap = lds;             // [64][64]
  float* Bp = lds + 64 * 64;   // [64][64]
  for (int kb = 0; kb < K; kb += 64) {
    __syncthreads();
    for (int it = 0; it < 4; ++it) {
      int i = tid + it * 256;
      int r = i >> 4, c4 = i & 15;
      *(float4*)&Ap[r * 64 + c4 * 4] =
          *(const float4*)&A[(long long)(m0 + r) * K + kb + c4 * 4];
      *(float4*)&Bp[r * 64 + c4 * 4] =
          *(const float4*)&Bm[(long long)(kb + r) * N + n0 + c4 * 4];
    }
    __syncthreads();
    for (int kk = 0; kk < 64; kk += 4) {
      v2f a = *(const v2f*)&Ap[(trow * 16 + lm) * 64 + kk + half * 2];
      int kr0 = (kk + half * 2) * 64, kr1 = kr0 + 64;
      v2f b0, b1;
      b0.x = Bp[kr0 + tc0 * 16 + lm];
      b0.y = Bp[kr1 + tc0 * 16 + lm];
      b1.x = Bp[kr0 + tc1 * 16 + lm];
      b1.y = Bp[kr1 + tc1 * 16 + lm];
      acc0 = wmma4(a, b0, acc0);
      acc1 = wmma4(a, b1, acc1);
    }
  }
#endif

  for (int i = 0; i < 8; ++i) {
    int row = m0 + trow * 16 + i + 8 * half;
    float v0 = acc0[i], v1 = acc1[i];
    if (relu) { v0 = fmaxf(v0, 0.f); v1 = fmaxf(v1, 0.f); }
    C[(long long)row * N + n0 + tc0 * 16 + lm] = v0;
    C[(long long)row * N + n0 + tc1 * 16 + lm] = v1;
  }
}

// ---------------------------------------------------------------------------
// Kernel 4: last_h[b] = av2[b, alias[b, sum(mask[b])-1]]
// ---------------------------------------------------------------------------
__global__ void __launch_bounds__(64) k_lasth(
    const float* __restrict__ av2, const float* __restrict__ mask,
    const int* __restrict__ alias_, float* __restrict__ last_h) {
  const int b = blockIdx.x, t = threadIdx.x;
  float s = 0.f;
  for (int i = 0; i < SS; ++i) s += mask[b * SS + i];
  int rm = (int)(s + 0.5f);
  rm = rm < 1 ? 1 : (rm > SS ? SS : rm);
  int last = alias_[b * SS + rm - 1];
  *(float4*)&last_h[b * DD + t * 4] =
      *(const float4*)&av2[((long long)b * SS + last) * DD + t * 4];
}

// ---------------------------------------------------------------------------
// Kernel 5: logits[512, 99999] = last_h @ emb[1:].T
// One TDM load stages 128 embedding rows (128KB, n-major) into LDS; rows past
// N-1 are zero-filled by TDM OOB (tensor_dim1 = remaining rows). 8 waves each
// own one 16-col N-tile and sweep all 32 M-tiles.
// ---------------------------------------------------------------------------
__global__ void __launch_bounds__(256) k_logits(
    const float* __restrict__ emb, const float* __restrict__ last_h,
    float* __restrict__ logits) {
  extern __shared__ float et[];  // [128][256] n-major
  const int nb  = blockIdx.x * 128;
  const int tid = threadIdx.x;
  const int w = tid >> 5, l = tid & 31, half = l >> 4, lm = l & 15;

#ifdef USE_TDM
  if (w == 0) {
    unsigned rows_left = (unsigned)(NM1 - nb);  // valid rows from emb row 1+nb
    tdm_load_2d(0u, &emb[(long long)(1 + nb) * DD],
                DD, 128, DD, DD, rows_left);
    __builtin_amdgcn_s_wait_tensorcnt(0);
  }
  __syncthreads();
#else
  for (int it = 0; it < 32; ++it) {
    int i  = tid + it * 256;
    int nn = i >> 6, c4 = i & 63;
    int gn = nb + nn;
    float4 v = make_float4(0.f, 0.f, 0.f, 0.f);
    if (gn < NM1)
      v = *(const float4*)&emb[((long long)(1 + gn)) * DD + c4 * 4];
    *(float4*)&et[nn * DD + c4 * 4] = v;
  }
  __syncthreads();
#endif

  const int nloc = w * 16;
  for (int mt = 0; mt < 32; ++mt) {
    v8f acc = {};
    for (int kk = 0; kk < DD; kk += 4) {
      v2f a = *(const v2f*)&last_h[(mt * 16 + lm) * DD + kk + half * 2];
      v2f b = *(const v2f*)&et[(nloc + lm) * DD + kk + half * 2];
      acc = wmma4(a, b, acc);
    }
    int gn = nb + nloc + lm;
    if (gn < NM1) {
      for (int i = 0; i < 8; ++i)
        logits[(long long)(mt * 16 + i + 8 * half) * NM1 + gn] = acc[i];
    }
  }
}

// ---------------------------------------------------------------------------
// Kernel 6: per-row log-softmax NLL:  rowloss[b] = lse(row) - row[tar[b]-1]
// Fixed-order LDS tree reductions -> deterministic.
// ---------------------------------------------------------------------------
__global__ void __launch_bounds__(256) k_rowloss(
    const float* __restrict__ logits, const int* __restrict__ tar,
    float* __restrict__ rowloss) {
  __shared__ float red[256];
  const int b = blockIdx.x, tid = threadIdx.x;
  const float* row = logits + (long long)b * NM1;

  float m = -3.402823466e38f;
  for (int i = tid; i < NM1; i += 256) m = fmaxf(m, row[i]);
  red[tid] = m;
  __syncthreads();
  for (int s = 128; s > 0; s >>= 1) {
    if (tid < s) red[tid] = fmaxf(red[tid], red[tid + s]);
    __syncthreads();
  }
  const float rowmax = red[0];
  __syncthreads();

  float acc = 0.f;
  for (int i = tid; i < NM1; i += 256) acc += expf(row[i] - rowmax);
  red[tid] = acc;
  __syncthreads();
  for (int s = 128; s > 0; s >>= 1) {
    if (tid < s) red[tid] += red[tid + s];
    __syncthreads();
  }
  if (tid == 0) {
    float lse = rowmax + logf(red[0]);
    int t = tar[b] - 1;
    rowloss[b] = lse - row[t];
  }
}

// Kernel 7: loss = mean(rowloss)
__global__ void __launch_bounds__(512) k_meanloss(
    const float* __restrict__ rowloss, float* __restrict__ out) {
  __shared__ float red[512];
  const int tid = threadIdx.x;
  red[tid] = rowloss[tid];
  __syncthreads();
  for (int s = 256; s > 0; s >>= 1) {
    if (tid < s) red[tid] += red[tid + s];
    __syncthreads();
  }
  if (tid == 0) out[0] = red[0] / (float)BB;
}

// ---------------------------------------------------------------------------
extern "C" void kernel_launch(void* const* d_in, const int* in_sizes, int n_in,
                              void* d_out, int out_size, void* d_ws,
                              size_t ws_size, hipStream_t stream) {
  const float* emb     = (const float*)d_in[0];  // [100000,256]
  const float* W1      = (const float*)d_in[1];  // [512,512]
  const float* W2      = (const float*)d_in[2];  // [512,256]
  const float* adj_in  = (const float*)d_in[3];  // [512,64,64]
  const float* adj_out = (const float*)d_in[4];  // [512,64,64]
  const float* mask    = (const float*)d_in[5];  // [512,64]
  const int*   item    = (const int*)d_in[6];    // [512,64]
  const int*   alias_  = (const int*)d_in[7];    // [512,64]
  const int*   tar     = (const int*)d_in[8];    // [512]

  float* out    = (float*)d_out;
  float* logits = out + 1;                       // [512,99999]

  char* ws = (char*)d_ws;
  float* av      = (float*)(ws);                         // [32768,512] 64MB
  float* H       = (float*)(ws + ((size_t)64 << 20));    // [32768,512] 64MB
  float* av2     = av;  // reuse av region once av is dead (GEMM2 reads only H)
  float* last_h  = (float*)(ws + ((size_t)128 << 20));   // [512,256]
  float* rowloss = last_h + (size_t)BB * DD;             // [512]

  // 1) gather + dual-adjacency aggregation -> av [B*S, 2D]
  k_adj_av<<<BB, 256, 64 * 1024, stream>>>(emb, adj_in, adj_out, item, av);

  // 2) H = relu(av @ W1)   [32768,512]   (64KB LDS: 2x(A+B) ping-pong panels)
  dim3 g1(TWOD / 64, (BB * SS) / 64);
  k_gemm_f32<<<g1, 256, 64 * 1024, stream>>>(av, W1, H, BB * SS, TWOD, TWOD, 1);

  // 3) av2 = H @ W2        [32768,256]
  dim3 g2(DD / 64, (BB * SS) / 64);
  k_gemm_f32<<<g2, 256, 64 * 1024, stream>>>(H, W2, av2, BB * SS, DD, TWOD, 0);

  // 4) last-node hidden
  k_lasth<<<BB, 64, 0, stream>>>(av2, mask, alias_, last_h);

  // 5) logits = last_h @ emb[1:].T  (written straight into d_out+1)
  k_logits<<<(NM1 + 127) / 128, 256, 128 * 1024, stream>>>(emb, last_h, logits);

  // 6) per-row NLL, 7) mean -> d_out[0]
  k_rowloss<<<BB, 256, 0, stream>>>(logits, tar, rowloss);
  k_meanloss<<<1, 512, 0, stream>>>(rowloss, out);
}